// ULDLoss_20813411516803
// MI455X (gfx1250) — compile-verified
//
#include <hip/hip_runtime.h>
#include <math.h>
#include <stdint.h>

// ---------------------------------------------------------------------------
// ULD loss (softmax -> group geometric-mean merge -> generalized JSD) fused.
// One workgroup per alignment group; both merged group log-vectors (2 x 125KB)
// live in CDNA5's 320KB LDS, so every logit row is read from HBM exactly once.
// ---------------------------------------------------------------------------

#define BLOCK    1024          // 32 wave32 waves per workgroup
#define NWAVES   (BLOCK / 32)
#define K4       8             // float4 chunks / thread  (supports V <= 32768)
#define GRID_MAX 4096          // upper bound on num_groups (G=1024 here)

#define LNEPS   (-18.420680743952367f)  // log(1e-8)
#define LNHALF  (-0.6931471805599453f)  // log(0.5)  (beta = 0.5)
#define BIGZ    (1e30f)

#ifndef __has_builtin
#define __has_builtin(x) 0
#endif

#if defined(__AMDGCN__) && __has_builtin(__builtin_amdgcn_global_load_async_to_lds_b128)
#define HAVE_ASYNC_LDS 1
#else
#define HAVE_ASYNC_LDS 0
#endif

#if HAVE_ASYNC_LDS
typedef int v4i __attribute__((vector_size(16)));
typedef __attribute__((address_space(1))) v4i* g_v4i_p;   // global (AS1) v4i*
typedef __attribute__((address_space(3))) v4i* l_v4i_p;   // LDS    (AS3) v4i*
#endif

// ---- (max, sumexp) pair combine: exact streaming logsumexp ----------------
__device__ __forceinline__ void lse_combine(float& m, float& s, float m2, float s2) {
  if (m2 == -INFINITY) return;
  if (m == -INFINITY) { m = m2; s = s2; return; }
  if (m2 > m) { s = s * expf(m - m2) + s2; m = m2; }
  else        { s = s + s2 * expf(m2 - m); }
}

// Block-wide logsumexp from per-thread (m, s). All threads must participate.
__device__ __forceinline__ float block_lse(float m, float s, float* red_m, float* red_s) {
  #pragma unroll
  for (int off = 16; off > 0; off >>= 1) {
    float m2 = __shfl_xor(m, off, 32);
    float s2 = __shfl_xor(s, off, 32);
    lse_combine(m, s, m2, s2);
  }
  const int wave = threadIdx.x >> 5;
  const int lane = threadIdx.x & 31;
  if (lane == 0) { red_m[wave] = m; red_s[wave] = s; }
  __syncthreads();
  if (wave == 0) {
    m = (lane < NWAVES) ? red_m[lane] : -INFINITY;
    s = (lane < NWAVES) ? red_s[lane] : 0.f;
    #pragma unroll
    for (int off = 16; off > 0; off >>= 1) {
      float m2 = __shfl_xor(m, off, 32);
      float s2 = __shfl_xor(s, off, 32);
      lse_combine(m, s, m2, s2);
    }
    if (lane == 0) { red_m[0] = m; red_s[0] = s; }
  }
  __syncthreads();
  const float r = red_m[0] + logf(red_s[0]);
  __syncthreads();                       // scratch reusable after this
  return r;
}

__device__ __forceinline__ float block_sum(float v, float* red) {
  #pragma unroll
  for (int off = 16; off > 0; off >>= 1) v += __shfl_xor(v, off, 32);
  const int wave = threadIdx.x >> 5;
  const int lane = threadIdx.x & 31;
  if (lane == 0) red[wave] = v;
  __syncthreads();
  if (wave == 0) {
    v = (lane < NWAVES) ? red[lane] : 0.f;
    #pragma unroll
    for (int off = 16; off > 0; off >>= 1) v += __shfl_xor(v, off, 32);
    if (lane == 0) red[0] = v;
  }
  __syncthreads();
  const float r = red[0];
  __syncthreads();
  return r;
}

// Accumulate glog[v] += max(row[v] - lse(row), ln eps) for rows [r0, r1).
// Row is staged in VGPRs (single HBM read); next row is prefetched.
__device__ __forceinline__ void accum_rows(const float* __restrict__ logits,
                                           int r0, int r1, int V,
                                           float* __restrict__ glog,
                                           float* red_m, float* red_s, int t) {
  const int  V4     = V >> 2;
  const bool staged = ((V & 3) == 0) && (V4 <= K4 * BLOCK);
  for (int r = r0; r < r1; ++r) {
    const float* row = logits + (size_t)r * (size_t)V;
    if (r + 1 < r1) __builtin_prefetch(row + (size_t)V + (size_t)t * 4, 0, 1);  // global_prefetch_b8
    if (staged) {
      const float4* row4 = (const float4*)row;
      float4 x[K4];
      float  m = -INFINITY;
      #pragma unroll
      for (int k = 0; k < K4; ++k) {
        const int i4 = t + k * BLOCK;
        if (i4 < V4) {
          x[k] = row4[i4];
          m = fmaxf(m, fmaxf(fmaxf(x[k].x, x[k].y), fmaxf(x[k].z, x[k].w)));
        }
      }
      float s = 0.f;
      #pragma unroll
      for (int k = 0; k < K4; ++k) {
        const int i4 = t + k * BLOCK;
        if (i4 < V4)
          s += expf(x[k].x - m) + expf(x[k].y - m) + expf(x[k].z - m) + expf(x[k].w - m);
      }
      const float lse = block_lse(m, s, red_m, red_s);
      #pragma unroll
      for (int k = 0; k < K4; ++k) {
        const int i4 = t + k * BLOCK;
        if (i4 < V4) {
          float4* gp = (float4*)(glog + 4 * i4);     // thread-private slots: no races
          float4  gv = *gp;
          gv.x += fmaxf(x[k].x - lse, LNEPS);
          gv.y += fmaxf(x[k].y - lse, LNEPS);
          gv.z += fmaxf(x[k].z - lse, LNEPS);
          gv.w += fmaxf(x[k].w - lse, LNEPS);
          *gp = gv;
        }
      }
    } else {  // generic fallback: streaming online lse, 2nd pass hits L2
      float m = -INFINITY, s = 0.f;
      for (int v = t; v < V; v += BLOCK) {
        const float x = row[v];
        if (x > m) { s = s * expf(m - x) + 1.f; m = x; }
        else       { s += expf(x - m); }
      }
      const float lse = block_lse(m, s, red_m, red_s);
      for (int v = t; v < V; v += BLOCK)
        glog[v] += fmaxf(row[v] - lse, LNEPS);
    }
  }
}

extern "C" __global__ void __launch_bounds__(BLOCK)
uld_zero(float* out) { if (threadIdx.x == 0) out[0] = 0.f; }

extern "C" __global__ void __launch_bounds__(BLOCK)
uld_group_jsd(const float* __restrict__ sLogit, const float* __restrict__ tLogit,
              const int* __restrict__ sGid, const int* __restrict__ tGid,
              const int* __restrict__ numGroups,
              const int S, const int T, const int V,
              float* __restrict__ out) {
  extern __shared__ float smem[];
  float* glog_s = smem;                  // V floats
  float* glog_t = smem + V;              // V floats
  float* red_m  = glog_t + V;            // NWAVES
  float* red_s  = red_m + NWAVES;        // NWAVES
  int*   range  = (int*)(red_s + NWAVES);// 4: sStart,sEnd,tStart,tEnd
  int*   gid_l  = range + 4;             // S + T ints (16B aligned for V%4==0)

  const int g = blockIdx.x;
  const int G = numGroups[0];
  if (g >= G) return;
  const int t = threadIdx.x;

  // ---- stage sorted group-id arrays into LDS (async DMA path on CDNA5) ----
#if HAVE_ASYNC_LDS
  if (((S | T) & 3) == 0) {
    const int c1 = S >> 2, nc = c1 + (T >> 2);
    for (int c = t; c < nc; c += BLOCK) {
      const int* src = (c < c1) ? (sGid + 4 * c) : (tGid + 4 * (c - c1));
      __builtin_amdgcn_global_load_async_to_lds_b128(
          (g_v4i_p)src, (l_v4i_p)(gid_l + 4 * c), 0, 0);
    }
  } else {
    for (int i = t; i < S; i += BLOCK) gid_l[i]     = sGid[i];
    for (int i = t; i < T; i += BLOCK) gid_l[S + i] = tGid[i];
  }
#else
  for (int i = t; i < S; i += BLOCK) gid_l[i]     = sGid[i];
  for (int i = t; i < T; i += BLOCK) gid_l[S + i] = tGid[i];
#endif

  // zero both group-log accumulators while the DMA is in flight
  for (int v = t; v < V; v += BLOCK) { glog_s[v] = 0.f; glog_t[v] = 0.f; }
  if (t < 2) { range[2 * t] = 0x7fffffff; range[2 * t + 1] = 0; }

#if HAVE_ASYNC_LDS
#if __has_builtin(__builtin_amdgcn_s_wait_asynccnt)
  __builtin_amdgcn_s_wait_asynccnt(0);
#else
  asm volatile("s_wait_asynccnt 0" ::: "memory");
#endif
#endif
  __syncthreads();

  // ---- parallel range find: ids are sorted so each group is contiguous ----
  for (int i = t; i < S; i += BLOCK)
    if (gid_l[i] == g) { atomicMin(&range[0], i); atomicMax(&range[1], i + 1); }
  for (int i = t; i < T; i += BLOCK)
    if (gid_l[S + i] == g) { atomicMin(&range[2], i); atomicMax(&range[3], i + 1); }
  __syncthreads();
  const int sS = range[0], sE = range[1], tS = range[2], tE = range[3];
  const int ns = (sE > sS) ? (sE - sS) : 0;
  const int nt = (tE > tS) ? (tE - tS) : 0;
  if ((ns | nt) == 0) return;  // empty-empty groups contribute exactly zero

  // ---- accumulate sum of clipped log-probs per side ----
  accum_rows(sLogit, sS, sE, V, glog_s, red_m, red_s, t);
  accum_rows(tLogit, tS, tE, V, glog_t, red_m, red_s, t);

  // ---- normalizers: Z = lse(glog) if size>1, 0 if size==1, +inf if empty ----
  float Zs, Zt;
  if (ns > 1) {
    float m = -INFINITY, s = 0.f;
    for (int v = t; v < V; v += BLOCK) {
      const float x = glog_s[v];
      if (x > m) { s = s * expf(m - x) + 1.f; m = x; } else { s += expf(x - m); }
    }
    Zs = block_lse(m, s, red_m, red_s);
  } else Zs = (ns == 1) ? 0.f : BIGZ;
  if (nt > 1) {
    float m = -INFINITY, s = 0.f;
    for (int v = t; v < V; v += BLOCK) {
      const float x = glog_t[v];
      if (x > m) { s = s * expf(m - x) + 1.f; m = x; } else { s += expf(x - m); }
    }
    Zt = block_lse(m, s, red_m, red_s);
  } else Zt = (nt == 1) ? 0.f : BIGZ;

  // ---- generalized JSD (beta = 0.5) on clipped merged log-probs ----
  float local = 0.f;
  for (int v = t; v < V; v += BLOCK) {
    const float sl = fmaxf(glog_s[v] - Zs, LNEPS);
    const float tl = fmaxf(glog_t[v] - Zt, LNEPS);
    const float a  = sl + LNHALF, b = tl + LNHALF;
    const float mx = fmaxf(a, b), mn = fminf(a, b);
    const float mix = mx + log1pf(expf(mn - mx));        // logaddexp
    local += 0.5f * (expf(tl) * (tl - mix) + expf(sl) * (sl - mix));
  }
  const float tot = block_sum(local, red_m);
  if (t == 0) atomicAdd(out, tot / (float)G);
}

extern "C" void kernel_launch(void* const* d_in, const int* in_sizes, int n_in,
                              void* d_out, int out_size, void* d_ws, size_t ws_size,
                              hipStream_t stream) {
  const float* sL = (const float*)d_in[0];
  const float* tL = (const float*)d_in[1];
  const int*   sG = (const int*)d_in[2];
  const int*   tG = (const int*)d_in[3];
  const int*   nG = (const int*)d_in[4];   // num_groups (device scalar)
  const int S = in_sizes[2];
  const int T = in_sizes[3];
  const int V = in_sizes[0] / S;           // 32000
  float* out = (float*)d_out;

  const size_t lds_bytes = (size_t)(2 * V + 2 * NWAVES) * sizeof(float)
                         + 4 * sizeof(int) + (size_t)(S + T) * sizeof(int);   // ~273 KB < 320 KB/WG

  (void)hipFuncSetAttribute((const void*)uld_group_jsd,
                            hipFuncAttributeMaxDynamicSharedMemorySize, (int)lds_bytes);

  uld_zero<<<1, BLOCK, 0, stream>>>(out);  // d_out is poisoned by the harness
  uld_group_jsd<<<GRID_MAX, BLOCK, lds_bytes, stream>>>(sL, tL, sG, tG, nG, S, T, V, out);
  (void)d_ws; (void)ws_size; (void)n_in; (void)out_size;
}